// GCN_test_2190433321522
// MI455X (gfx1250) — compile-verified
//
#include <hip/hip_runtime.h>
#include <hip/hip_bf16.h>
#include <stdint.h>

#define NFEAT  256
#define NHID   128
#define NCLASS  64

typedef float v2f __attribute__((ext_vector_type(2)));
typedef float v8f __attribute__((ext_vector_type(8)));

// ---------------------------------------------------------------------------
// WMMA f32 GEMM:  H[M, NT*16] = X[M, K] @ W[NT*16, K]^T
// One wave32 per 16-row M-tile; the wave iterates all NT column tiles so X is
// read exactly once.  V_WMMA_F32_16X16X4_F32 layouts (ISA 7.12.2):
//   A 16x4 : lane l -> row l&15,  a = {X[m][k0+2*hi], X[m][k0+2*hi+1]}
//   B 4x16 : lane l -> col l&15,  b = {W[n][k0+2*hi], W[n][k0+2*hi+1]}
//   D 16x16: vgpr i -> row i+8*hi, col l&15
// ---------------------------------------------------------------------------
template<int K, int NT>
__global__ void __launch_bounds__(256)
wmma_gemm_xwt(const float* __restrict__ X, const float* __restrict__ W,
              float* __restrict__ H, int mtiles)
{
  const int wave = (int)((blockIdx.x * blockDim.x + threadIdx.x) >> 5);
  if (wave >= mtiles) return;                 // wave-uniform: EXEC stays all-1s
  const int lane = (int)(threadIdx.x & 31);
  const int r    = lane & 15;
  const int kk   = (lane >> 4) * 2;           // 0 or 2

  const float* xrow = X + (size_t)(wave * 16 + r) * K + kk;

  v8f acc[NT];
#pragma unroll
  for (int t = 0; t < NT; ++t)
#pragma unroll
    for (int i = 0; i < 8; ++i) acc[t][i] = 0.0f;

#pragma unroll 4
  for (int k0 = 0; k0 < K; k0 += 4) {
    v2f a = *(const v2f*)(xrow + k0);
#pragma unroll
    for (int t = 0; t < NT; ++t) {
      v2f b = *(const v2f*)(W + (size_t)(t * 16 + r) * K + kk + k0);
      acc[t] = __builtin_amdgcn_wmma_f32_16x16x4_f32(
          /*neg_a=*/false, a, /*neg_b=*/false, b,
          /*c_mod=*/(short)0, acc[t], /*reuse_a=*/false, /*reuse_b=*/false);
    }
  }

  const int Nc  = NT * 16;
  const int rhi = (lane >> 4) * 8;
  float* hb = H + (size_t)(wave * 16) * Nc;
#pragma unroll
  for (int t = 0; t < NT; ++t)
#pragma unroll
    for (int i = 0; i < 8; ++i)
      hb[(size_t)(i + rhi) * Nc + t * 16 + r] = acc[t][i];
}

// ---------------------------------------------------------------------------
// CSR-by-destination build (integer atomics only: 1.6M instead of 307M fp).
// ---------------------------------------------------------------------------
__global__ void hist_kernel(const long long* __restrict__ ei, int E,
                            int* __restrict__ cnt)
{
  int e = blockIdx.x * blockDim.x + threadIdx.x;
  if (e < E) atomicAdd(&cnt[(int)ei[(size_t)E + e]], 1);   // row 1 = dst
}

// Single-block chunked exclusive scan over node degrees -> rowptr + cursor.
__global__ void scan_kernel(int* __restrict__ cnt_cursor,   // in: counts, out: start offsets
                            int* __restrict__ rowptr, int n)
{
  __shared__ int buf[1024];
  __shared__ int carry;
  const int tid = threadIdx.x;
  if (tid == 0) carry = 0;
  __syncthreads();
  for (int base = 0; base < n; base += 1024) {
    int i = base + tid;
    int v = (i < n) ? cnt_cursor[i] : 0;
    buf[tid] = v;
    __syncthreads();
    for (int off = 1; off < 1024; off <<= 1) {
      int t = (tid >= off) ? buf[tid - off] : 0;
      __syncthreads();
      buf[tid] += t;
      __syncthreads();
    }
    int excl = buf[tid] - v;
    int c = carry;
    __syncthreads();
    if (i < n) { rowptr[i] = c + excl; cnt_cursor[i] = c + excl; }
    if (tid == 1023) carry = c + buf[1023];
    __syncthreads();
  }
  if (tid == 0) rowptr[n] = carry;
}

__global__ void fill_kernel(const long long* __restrict__ ei, int E,
                            int* __restrict__ cursor, int* __restrict__ eid)
{
  int e = blockIdx.x * blockDim.x + threadIdx.x;
  if (e < E) {
    int pos = atomicAdd(&cursor[(int)ei[(size_t)E + e]], 1);
    eid[pos] = e;
  }
}

// ---------------------------------------------------------------------------
// Gather-reduce: one wave32 per node, lane owns F/32 features. H tables fit in
// the 192 MB L2, so these random gathers are L2-resident. Optional fused ReLU.
// ---------------------------------------------------------------------------
template<int F, bool RELU>
__global__ void __launch_bounds__(256)
gather_aggregate(const float* __restrict__ Hsrc,
                 const long long* __restrict__ ei,     // row 0 = src
                 const float* __restrict__ ew,
                 const int* __restrict__ rowptr,
                 const int* __restrict__ eid,
                 float* __restrict__ outp, int nnodes)
{
  constexpr int V = F / 32;
  const int wave = (int)((blockIdx.x * blockDim.x + threadIdx.x) >> 5);
  if (wave >= nnodes) return;
  const int lane = (int)(threadIdx.x & 31);
  const int s = rowptr[wave];
  const int t = rowptr[wave + 1];

  float acc[V];
#pragma unroll
  for (int v = 0; v < V; ++v) acc[v] = 0.0f;

  const float* base = Hsrc + lane * V;
  for (int j = s; j < t; ++j) {
    const int e        = eid[j];
    const long long sn = ei[e];
    const float w      = ew[e];
    const float* hp = base + (size_t)sn * F;
    if constexpr (V == 4) {
      const float4 h = *(const float4*)hp;
      acc[0] += w * h.x; acc[1] += w * h.y;
      acc[2] += w * h.z; acc[3] += w * h.w;
    } else {
      const float2 h = *(const float2*)hp;
      acc[0] += w * h.x; acc[1] += w * h.y;
    }
  }

  float* op = outp + (size_t)wave * F + lane * V;
#pragma unroll
  for (int v = 0; v < V; ++v) {
    float x = acc[v];
    if (RELU) x = fmaxf(x, 0.0f);
    op[v] = x;
  }
}

// ---------------------------------------------------------------------------
extern "C" void kernel_launch(void* const* d_in, const int* in_sizes, int n_in,
                              void* d_out, int out_size, void* d_ws, size_t ws_size,
                              hipStream_t stream)
{
  const float*     x   = (const float*)d_in[0];
  const long long* ei1 = (const long long*)d_in[1];   // int64 [2,E]
  const long long* ei2 = (const long long*)d_in[2];
  const float*     ew1 = (const float*)d_in[3];
  const float*     ew2 = (const float*)d_in[4];
  const float*     W1  = (const float*)d_in[5];       // [128,256]
  const float*     W2  = (const float*)d_in[6];       // [64,128]

  const int N    = in_sizes[0] / NFEAT;               // 100000 (multiple of 16)
  const int E1   = in_sizes[1] / 2;
  const int E2   = in_sizes[2] / 2;
  const int Emax = E1 > E2 ? E1 : E2;

  // workspace layout
  char* wsb = (char*)d_ws;
  size_t off = 0;
  float* H1 = (float*)(wsb + off); off += (size_t)N * NHID * sizeof(float);
  float* A1 = (float*)(wsb + off); off += (size_t)N * NHID * sizeof(float);
  int* rowptr = (int*)(wsb + off); off += (size_t)(N + 1) * sizeof(int);
  int* cursor = (int*)(wsb + off); off += (size_t)N * sizeof(int);
  int* eid    = (int*)(wsb + off); off += (size_t)Emax * sizeof(int);
  float* H2 = H1;   // H1 is dead after layer-1 aggregation; reuse for H2

  const int mtiles       = N / 16;
  const int gemm_blocks  = (mtiles * 32 + 255) / 256;
  const int nodew_blocks = (N * 32 + 255) / 256;
  (void)n_in; (void)out_size; (void)ws_size;

  // ----------------- layer 1: H1 = X @ W1^T -----------------
  wmma_gemm_xwt<NFEAT, NHID / 16><<<gemm_blocks, 256, 0, stream>>>(x, W1, H1, mtiles);

  hipMemsetAsync(cursor, 0, (size_t)N * sizeof(int), stream);
  hist_kernel<<<(E1 + 255) / 256, 256, 0, stream>>>(ei1, E1, cursor);
  scan_kernel<<<1, 1024, 0, stream>>>(cursor, rowptr, N);
  fill_kernel<<<(E1 + 255) / 256, 256, 0, stream>>>(ei1, E1, cursor, eid);
  gather_aggregate<NHID, true><<<nodew_blocks, 256, 0, stream>>>(
      H1, ei1, ew1, rowptr, eid, A1, N);              // fused ReLU

  // ----------------- layer 2: H2 = A1 @ W2^T ----------------
  wmma_gemm_xwt<NHID, NCLASS / 16><<<gemm_blocks, 256, 0, stream>>>(A1, W2, H2, mtiles);

  hipMemsetAsync(cursor, 0, (size_t)N * sizeof(int), stream);
  hist_kernel<<<(E2 + 255) / 256, 256, 0, stream>>>(ei2, E2, cursor);
  scan_kernel<<<1, 1024, 0, stream>>>(cursor, rowptr, N);
  fill_kernel<<<(E2 + 255) / 256, 256, 0, stream>>>(ei2, E2, cursor, eid);
  gather_aggregate<NCLASS, false><<<nodew_blocks, 256, 0, stream>>>(
      H2, ei2, ew2, rowptr, eid, (float*)d_out, N);   // writes every output
}